// VocabParallelEmbedding_35862976921833
// MI455X (gfx1250) — compile-verified
//
#include <hip/hip_runtime.h>
#include <hip/hip_bf16.h>

// Embedding gather: out[token, :] = weight[idx[token], :]
//   idx:    (B*S,)        int32   (B=4, S=4096 -> 16384 tokens)
//   weight: (50257, 1024) float32
//   out:    (B*S, 1024)   float32
//
// Pure bandwidth problem: 64 MiB gather-read + 64 MiB streaming-write
// (~5.7 us at 23.3 TB/s). No FLOPs -> WMMA is irrelevant; the gfx1250
// feature that matters is the async LDS copy engine:
//   global_load_async_to_lds_b128   (HBM -> LDS, per-lane, ASYNCcnt)
//   s_wait_asynccnt 0
//   global_store_async_from_lds_b128 th:TH_STORE_NT (LDS -> HBM)
//
// Layout: ONE WAVE PER TOKEN (16384 waves total, vs 131k in v1 -> cuts
// wave-launch overhead 8x). Each wave moves its 4 KB row as 8 x 512 B
// async b128 issues per direction; the instruction IOFFSET is applied to
// both global and LDS addresses, so the 8 chunks share one address setup.
// Each lane stages only through its own LDS bytes -> no workgroup barrier,
// just the wave-local ASYNCcnt wait. Output stores use TH_STORE_NT so the
// write-once output doesn't evict reusable weight rows from the 192 MB L2.

#define HIDDEN      1024
#define ROW_BYTES   (HIDDEN * 4)     // 4096 B per row
#define WAVES_PB    8
#define THREADS     (WAVES_PB * 32)  // 256
#define CHUNK_BYTES 512              // 32 lanes * 16 B
#define CHUNKS      (ROW_BYTES / CHUNK_BYTES)  // 8

__global__ __launch_bounds__(THREADS) void VocabParallelEmbedding_gather_async(
    const int* __restrict__ idx,
    const float* __restrict__ weight,
    float* __restrict__ out,
    int ntok)
{
    __shared__ __align__(16) unsigned char lds_buf[WAVES_PB * ROW_BYTES]; // 32 KB

    const unsigned wave = threadIdx.x >> 5;
    const unsigned lane = threadIdx.x & 31;
    const unsigned token = blockIdx.x * WAVES_PB + wave;
    if (token >= (unsigned)ntok) return;

    const int row = idx[token];   // wave-uniform value (same addr in all lanes)

    const unsigned char* gsrc =
        (const unsigned char*)weight + (size_t)row * ROW_BYTES + (size_t)lane * 16;
    unsigned char* gdst =
        (unsigned char*)out + (size_t)token * ROW_BYTES + (size_t)lane * 16;

    // LDS byte offset of this lane's slot (generic ptr low 32 bits == DS addr).
    unsigned lds_off = (unsigned)(unsigned long long)
        (lds_buf + (size_t)wave * ROW_BYTES + (size_t)lane * 16);

    // HBM -> LDS: 8 async b128 issues cover the 4 KB row.
    #pragma unroll
    for (int k = 0; k < CHUNKS; ++k) {
        asm volatile("global_load_async_to_lds_b128 %0, %1, off offset:%c2"
                     :: "v"(lds_off), "v"(gsrc), "i"(k * CHUNK_BYTES)
                     : "memory");
    }

    // Wave-local: all of this wave's loads have landed in LDS.
    asm volatile("s_wait_asynccnt 0x0" ::: "memory");

    // LDS -> HBM: non-temporal stores (output is write-once, never re-read).
    #pragma unroll
    for (int k = 0; k < CHUNKS; ++k) {
        asm volatile("global_store_async_from_lds_b128 %0, %1, off offset:%c2 th:TH_STORE_NT"
                     :: "v"(gdst), "v"(lds_off), "i"(k * CHUNK_BYTES)
                     : "memory");
    }
    // s_endpgm performs an implicit wait-idle, covering the pending stores.
}

extern "C" void kernel_launch(void* const* d_in, const int* in_sizes, int n_in,
                              void* d_out, int out_size, void* d_ws, size_t ws_size,
                              hipStream_t stream) {
    const int*   idx    = (const int*)d_in[0];     // (B*S,) int32
    const float* weight = (const float*)d_in[1];   // (VOCAB, HIDDEN) f32
    float*       out    = (float*)d_out;           // (B*S, HIDDEN) f32

    const int tokens = in_sizes[0];                // 16384
    const int blocks = (tokens + WAVES_PB - 1) / WAVES_PB;  // 2048

    VocabParallelEmbedding_gather_async<<<blocks, THREADS, 0, stream>>>(
        idx, weight, out, tokens);
}